// FusionModule_4217657884733
// MI455X (gfx1250) — compile-verified
//
#include <hip/hip_runtime.h>
#include <hip/hip_bf16.h>

typedef __attribute__((ext_vector_type(2))) float v2f;
typedef __attribute__((ext_vector_type(4))) float v4f;
typedef __attribute__((ext_vector_type(8))) float v8f;

#define B_  32
#define L_  1024
#define DG_ 1024
#define H_  1024

// ---------------------------------------------------------------------------
// Kernel A: gate[B, L] = sigmoid(query[B,H] @ W[H,L] + bias[L])
// One wave per 16x16 output tile; V_WMMA_F32_16X16X4_F32, K stepped by 4.
// Tiles: M: 32/16 = 2, N: 1024/16 = 64  -> 128 waves total.
// ---------------------------------------------------------------------------
__global__ __launch_bounds__(128) void gate_gemm_wmma(
    const float* __restrict__ query,   // [B, H] row-major
    const float* __restrict__ W,       // [H, L] row-major
    const float* __restrict__ bias,    // [L]
    float* __restrict__ gate)          // [B, L]
{
  const int wave = threadIdx.x >> 5;           // 0..3
  const int lane = threadIdx.x & 31;
  const int tile = blockIdx.x * 4 + wave;      // 0..127
  const int n_tile = tile & 63;                // 64 N tiles
  const int m_tile = tile >> 6;                // 2 M tiles
  const int lh = lane >> 4;                    // lane half: 0 or 1
  const int mn = lane & 15;                    // row/col index within tile

  // A layout (16x4 f32, ISA 7.12.2): lane half selects K+{0,1} vs K+{2,3},
  // vgpr selects within the pair  -> K = vgpr + 2*lh
  const float* qrow = query + (size_t)(m_tile * 16 + mn) * H_;
  // B layout (4x16 f32): row K striped across 16 lanes of one vgpr;
  // vgpr0 holds K = 2*lh, vgpr1 holds K = 2*lh + 1 (mirrors A).
  const int n0 = n_tile * 16 + mn;

  v8f acc = {};
  #pragma unroll 4
  for (int k = 0; k < H_; k += 4) {
    v2f a, b;
    const int ka = k + 2 * lh;
    a.x = qrow[ka + 0];
    a.y = qrow[ka + 1];
    b.x = W[(size_t)(ka + 0) * L_ + n0];
    b.y = W[(size_t)(ka + 1) * L_ + n0];
    // (neg_a, A, neg_b, B, c_mod, C, reuse_a, reuse_b)
    acc = __builtin_amdgcn_wmma_f32_16x16x4_f32(
        false, a, false, b, (short)0, acc, false, false);
  }

  // C/D layout: vgpr r, lanes0-15 -> M=r, lanes16-31 -> M=r+8; N = lane&15.
  const float bn = bias[n0];
  #pragma unroll
  for (int r = 0; r < 8; ++r) {
    const int m = m_tile * 16 + r + 8 * lh;
    const float x = acc[r] + bn;
    gate[(size_t)m * L_ + n0] = 1.0f / (1.0f + __expf(-x));
  }
}

// ---------------------------------------------------------------------------
// Kernel B: pure bandwidth.  out[b,l,0:DG] = gate[b,l] * graph[b,l,:]
//                            out[b,l,DG:DG+H] = query[b,:]
// One block per (b,l) row; 256 threads x float4 covers DG=1024 and H=1024.
// Non-temporal on the 384 MB streamed; gate/query ride the 192 MB L2.
// ---------------------------------------------------------------------------
__global__ __launch_bounds__(256) void fuse_stream(
    const float* __restrict__ graph,   // [B, L, DG]
    const float* __restrict__ query,   // [B, H]
    const float* __restrict__ gate,    // [B, L]
    float* __restrict__ out)           // [B, L, DG+H]
{
  const int row = blockIdx.x;          // b*L + l
  const int b   = row >> 10;           // L_ == 1024
  const int t   = threadIdx.x;         // 0..255

  const float g = gate[row];

  const v4f* gr = (const v4f*)(graph + (size_t)row * DG_);
  const v4f* qr = (const v4f*)(query + (size_t)b * H_);
  v4f* o = (v4f*)(out + (size_t)row * (DG_ + H_));

  v4f v = __builtin_nontemporal_load(gr + t);
  v4f s;
  s.x = g * v.x; s.y = g * v.y; s.z = g * v.z; s.w = g * v.w;
  __builtin_nontemporal_store(s, o + t);

  v4f q = qr[t];                       // hot in L2 (128 KB total)
  __builtin_nontemporal_store(q, o + 256 + t);
}

// ---------------------------------------------------------------------------
extern "C" void kernel_launch(void* const* d_in, const int* in_sizes, int n_in,
                              void* d_out, int out_size, void* d_ws, size_t ws_size,
                              hipStream_t stream) {
  const float* graph = (const float*)d_in[0];  // [B, L, DG]
  const float* query = (const float*)d_in[1];  // [B, H]
  const float* W     = (const float*)d_in[2];  // [H, L]
  const float* bias  = (const float*)d_in[3];  // [L]
  float* out  = (float*)d_out;                 // [B, L, DG+H]
  float* gate = (float*)d_ws;                  // B*L floats = 128 KB scratch

  gate_gemm_wmma<<<32, 128, 0, stream>>>(query, W, bias, gate);
  fuse_stream<<<B_ * L_, 256, 0, stream>>>(graph, query, gate, out);
}